// NetVLAD_25271587570185
// MI455X (gfx1250) — compile-verified
//
#include <hip/hip_runtime.h>
#include <math.h>

// ---------------------------------------------------------------------------
// NetVLAD fused kernel for MI455X (gfx1250), fp32 WMMA (V_WMMA_F32_16X16X4_F32)
// N=64, C=64, HW=4096, K=128. out = [N, K*C] fp32.
// All WMMA fragments are single aligned ds_load_b64s (pair-interleaved B tiles).
// ---------------------------------------------------------------------------

typedef __attribute__((ext_vector_type(2))) float v2f;
typedef __attribute__((ext_vector_type(8))) float v8f;

#if defined(__HIP_DEVICE_COMPILE__) && !__has_builtin(__builtin_amdgcn_wmma_f32_16x16x4_f32)
#error "wmma_f32_16x16x4_f32 builtin not available on this toolchain (device pass)"
#endif

#define WMMA_F32(a, b, c) \
    __builtin_amdgcn_wmma_f32_16x16x4_f32(false, (a), false, (b), (short)0, (c), false, false)

namespace {
constexpr int Ndim = 64;
constexpr int Cdim = 64;
constexpr int HW   = 4096;
constexpr int Kdim = 128;
constexpr int TILE = 64;        // pixels per LDS tile
constexpr int SLICES = 8;       // L-slices per image
constexpr int TILES_PER_SLICE = (HW / SLICES) / TILE;  // 8
constexpr int XN_STRIDE = Cdim + 2;   // 66: pad kills bank conflicts, keeps 8B align
constexpr int SA_STRIDE = TILE + 2;   // 66
}

// ===========================================================================
// Kernel 1: per-slice normalize + logits GEMM + softmax + VLAD GEMM,
// partial vlad (with centroid term already subtracted) atomically added to out.
// grid = (SLICES, N), block = 256 (8 wave32)
// ===========================================================================
__global__ __launch_bounds__(256) void netvlad_main(
    const float* __restrict__ x,          // [N, C, HW]
    const float* __restrict__ centroids,  // [K, C]
    const float* __restrict__ conv_w,     // [K, C]
    const float* __restrict__ conv_b,     // [K]
    float* __restrict__ out)              // [N, K, C] accumulated
{
    __shared__ float xn [TILE * XN_STRIDE];    // pixel-major normalized tile (GEMM1 A)
    __shared__ float xnP[(TILE / 2) * Cdim * 2]; // pair-interleaved over l (GEMM2 B)
    __shared__ float WtP[(Cdim / 2) * Kdim * 2]; // conv_w pair-interleaved over c (GEMM1 B)
    __shared__ float saT[Kdim * SA_STRIDE];    // logits, then softmax [k][l] (GEMM2 A)
    __shared__ float bias[Kdim];
    __shared__ float ssum[Kdim];               // per-slice sum_l sa[k][l]
    __shared__ float scale_px[TILE];
    __shared__ float red[4 * TILE];
    __shared__ float maxv[TILE];
    __shared__ float denom[TILE];

    const int t    = threadIdx.x;
    const int n    = blockIdx.y;
    const int sl   = blockIdx.x;
    const int wave = t >> 5;
    const int lane = t & 31;
    const int g    = lane >> 4;   // half-wave: 0 or 1
    const int hp   = lane & 15;   // position within 16

    // --- load conv_w pair-interleaved + bias, zero slice ssum -------------
    for (int i = t; i < Kdim * Cdim; i += 256) {
        int k = i >> 6, c = i & 63;                  // consecutive t -> consecutive c
        WtP[((c >> 1) * Kdim + k) * 2 + (c & 1)] = conv_w[i];
    }
    if (t < Kdim) { bias[t] = conv_b[t]; ssum[t] = 0.0f; }
    __syncthreads();

    // GEMM2 accumulators: wave owns k-rows [16*wave, 16*wave+16), all 4 c-tiles
    v8f acc0 = {}, acc1 = {}, acc2 = {}, acc3 = {};

    const int l_base = sl * (HW / SLICES);

    for (int tile = 0; tile < TILES_PER_SLICE; ++tile) {
        const int l0 = l_base + tile * TILE;

        // ---- phase 1: load x tile (transpose [c][l] -> [l][c]) -----------
        for (int i = t; i < TILE * Cdim; i += 256) {
            int c = i >> 6, ll = i & 63;             // coalesced global: consecutive ll
            xn[ll * XN_STRIDE + c] =
                x[((size_t)n * Cdim + c) * HW + (l0 + ll)];
        }
        // prefetch next tile's 128B cachelines (global_prefetch_b8)
        if (tile + 1 < TILES_PER_SLICE && t < 128) {
            const int c = t >> 1, half = t & 1;
            __builtin_prefetch(&x[((size_t)n * Cdim + c) * HW + (l0 + TILE) + half * 32], 0, 1);
        }
        __syncthreads();
        if (t < TILE) {
            float s = 0.0f;
            #pragma unroll 8
            for (int c = 0; c < Cdim; ++c) { float v = xn[t * XN_STRIDE + c]; s += v * v; }
            scale_px[t] = 1.0f / fmaxf(sqrtf(s), 1e-12f);
        }
        __syncthreads();
        for (int i = t; i < TILE * Cdim; i += 256) {
            int ll = i >> 6, c = i & 63;
            float v = xn[ll * XN_STRIDE + c] * scale_px[ll];
            xn[ll * XN_STRIDE + c] = v;
            xnP[((ll >> 1) * Cdim + c) * 2 + (ll & 1)] = v;   // GEMM2 B copy
        }
        __syncthreads();

        // ---- phase 2: GEMM1 logits = xn(16px x 64c) @ W(64c x 128k) ------
        {
            const int m_t = wave & 3;     // pixel tile (16 pixels)
            const int kg  = wave >> 2;    // 0/1 -> which half of k
            const int m   = m_t * 16 + hp;

            v8f d0 = {}, d1 = {}, d2 = {}, d3 = {};
            #pragma unroll
            for (int cc = 0; cc < 16; ++cc) {
                const int c0 = cc * 4;
                const int c2 = 2 * cc + g;          // pair index c0/2 + g
                v2f a = *(const v2f*)&xn[m * XN_STRIDE + c0 + 2 * g];
                #pragma unroll
                for (int j = 0; j < 4; ++j) {
                    const int k = kg * 64 + j * 16 + hp;
                    v2f b = *(const v2f*)&WtP[(c2 * Kdim + k) * 2];
                    if (j == 0) d0 = WMMA_F32(a, b, d0);
                    if (j == 1) d1 = WMMA_F32(a, b, d1);
                    if (j == 2) d2 = WMMA_F32(a, b, d2);
                    if (j == 3) d3 = WMMA_F32(a, b, d3);
                }
            }
            // scatter logits + bias into saT[k][l]
            #pragma unroll
            for (int j = 0; j < 4; ++j) {
                const int k = kg * 64 + j * 16 + hp;
                const float bk = bias[k];
                #pragma unroll
                for (int r = 0; r < 8; ++r) {
                    const int l = m_t * 16 + r + 8 * g;
                    float dv = (j == 0) ? d0[r] : (j == 1) ? d1[r] : (j == 2) ? d2[r] : d3[r];
                    saT[k * SA_STRIDE + l] = dv + bk;
                }
            }
        }
        __syncthreads();

        // ---- phase 3: softmax over k (128) per pixel ---------------------
        {
            const int l = t & 63;
            const int q = t >> 6;      // quarter of k-range
            float mx = -3.402823e38f;
            for (int k = 32 * q; k < 32 * q + 32; ++k)
                mx = fmaxf(mx, saT[k * SA_STRIDE + l]);
            red[q * TILE + l] = mx;
            __syncthreads();
            if (t < TILE)
                maxv[t] = fmaxf(fmaxf(red[t], red[TILE + t]),
                                fmaxf(red[2 * TILE + t], red[3 * TILE + t]));
            __syncthreads();
            const float mall = maxv[l];
            float s = 0.0f;
            for (int k = 32 * q; k < 32 * q + 32; ++k) {
                float e = __expf(saT[k * SA_STRIDE + l] - mall);
                saT[k * SA_STRIDE + l] = e;
                s += e;
            }
            red[q * TILE + l] = s;
            __syncthreads();
            if (t < TILE)
                denom[t] = 1.0f / (red[t] + red[TILE + t] + red[2 * TILE + t] + red[3 * TILE + t]);
            __syncthreads();
            const float inv = denom[l];
            for (int k = 32 * q; k < 32 * q + 32; ++k)
                saT[k * SA_STRIDE + l] *= inv;
        }
        __syncthreads();
        // slice ssum[k] accumulation
        if (t < Kdim) {
            float s2 = 0.0f;
            #pragma unroll 8
            for (int l2 = 0; l2 < TILE; ++l2) s2 += saT[t * SA_STRIDE + l2];
            ssum[t] += s2;
        }
        __syncthreads();

        // ---- phase 4: GEMM2 vlad(16k x 64c) += sa(16k x 64l) @ xn(64l x 64c)
        {
            const int kr = wave * 16 + hp;   // A-matrix row (k)
            #pragma unroll
            for (int s4 = 0; s4 < 16; ++s4) {
                const int lc = s4 * 4;
                const int p  = 2 * s4 + g;           // pair index lc/2 + g
                v2f a = *(const v2f*)&saT[kr * SA_STRIDE + lc + 2 * g];
                #pragma unroll
                for (int ct = 0; ct < 4; ++ct) {
                    const int c = ct * 16 + hp;
                    v2f b = *(const v2f*)&xnP[(p * Cdim + c) * 2];
                    if (ct == 0) acc0 = WMMA_F32(a, b, acc0);
                    if (ct == 1) acc1 = WMMA_F32(a, b, acc1);
                    if (ct == 2) acc2 = WMMA_F32(a, b, acc2);
                    if (ct == 3) acc3 = WMMA_F32(a, b, acc3);
                }
            }
        }
        __syncthreads();
    }

    // --- epilogue: subtract slice ssum * centroids, atomic add to out -----
    #pragma unroll
    for (int ct = 0; ct < 4; ++ct) {
        #pragma unroll
        for (int r = 0; r < 8; ++r) {
            const int k = wave * 16 + r + 8 * g;
            const int c = ct * 16 + hp;
            float av = (ct == 0) ? acc0[r] : (ct == 1) ? acc1[r] : (ct == 2) ? acc2[r] : acc3[r];
            float v = av - ssum[k] * centroids[k * Cdim + c];
            unsafeAtomicAdd(&out[(size_t)n * (Kdim * Cdim) + k * Cdim + c], v);
        }
    }
}

// ===========================================================================
// Kernel 2: in-place intra-cluster L2 norm + global L2 norm per image.
// grid = N, block = 256
// ===========================================================================
__global__ __launch_bounds__(256) void netvlad_finalize(float* __restrict__ out)
{
    __shared__ float vbuf[Kdim * Cdim];   // 32 KB
    __shared__ float pss[256];
    __shared__ float rscale[Kdim];
    __shared__ float gred[64];
    __shared__ float gscale_sh;

    const int n = blockIdx.x;
    const int t = threadIdx.x;
    float* img = out + (size_t)n * (Kdim * Cdim);

    for (int i = t; i < Kdim * Cdim; i += 256) vbuf[i] = img[i];
    __syncthreads();

    // per-cluster sum of squares: 2 threads per k, 32 c each
    {
        const int k = t >> 1, half = t & 1;
        float p = 0.0f;
        #pragma unroll 8
        for (int c = 32 * half; c < 32 * half + 32; ++c) {
            float v = vbuf[k * Cdim + c];
            p += v * v;
        }
        pss[t] = p;
    }
    __syncthreads();
    if (t < Kdim) {
        float s = pss[2 * t] + pss[2 * t + 1];
        float sc = 1.0f / fmaxf(sqrtf(s), 1e-12f);
        rscale[t] = sc;
        pss[t] = s * sc * sc;    // row energy after intra-norm
    }
    __syncthreads();
    if (t < 64) gred[t] = pss[t] + pss[t + 64];
    __syncthreads();
    if (t < 16) gred[t] = gred[t] + gred[t + 16] + gred[t + 32] + gred[t + 48];
    __syncthreads();
    if (t == 0) {
        float tot = 0.0f;
        #pragma unroll
        for (int i = 0; i < 16; ++i) tot += gred[i];
        gscale_sh = 1.0f / fmaxf(sqrtf(tot), 1e-12f);
    }
    __syncthreads();

    const float gs = gscale_sh;
    for (int i = t; i < Kdim * Cdim; i += 256) {
        const int k = i >> 6;
        img[i] = vbuf[i] * rscale[k] * gs;
    }
}

// ===========================================================================
extern "C" void kernel_launch(void* const* d_in, const int* in_sizes, int n_in,
                              void* d_out, int out_size, void* d_ws, size_t ws_size,
                              hipStream_t stream)
{
    const float* x         = (const float*)d_in[0];
    const float* centroids = (const float*)d_in[1];
    const float* conv_w    = (const float*)d_in[2];
    const float* conv_b    = (const float*)d_in[3];
    float* out = (float*)d_out;

    // out is the vlad accumulator (N*K*C floats == out_size): zero it each call
    (void)hipMemsetAsync(out, 0, (size_t)out_size * sizeof(float), stream);

    dim3 grid(SLICES, Ndim);
    netvlad_main<<<grid, 256, 0, stream>>>(x, centroids, conv_w, conv_b, out);
    netvlad_finalize<<<Ndim, 256, 0, stream>>>(out);
}